// FrequencySegmentedMHFConv2D_38199439131366
// MI455X (gfx1250) — compile-verified
//
#include <hip/hip_runtime.h>
#include <math.h>
#include <stdint.h>

typedef __attribute__((ext_vector_type(2))) float v2f;
typedef __attribute__((ext_vector_type(8))) float v8f;

#define BATCH   32
#define C_IN    64
#define C_OUT   64
#define H_DIM   128
#define W_DIM   128
#define MPX_    16
#define WY_     65          // W/2 + 1
#define NROWS_  64          // n_heads * mpx  (x-frequency rows actually used)
#define NPTS_   (NROWS_ * WY_)   // 4160 frequency points
#define INV_SQRT_128 0.08838834764831845f

// LDS staging layout for the spectral kernel (float indices).
// Row strides chosen so S % 32 == 8  =>  2*S % 64 == 16: the two lane-halves
// (reading rows i0 and i0+2) hit disjoint 16-bank sets -> conflict-free.
#define SX_     40                          // padded stride for X rows (32 used)
#define SW_     72                          // padded stride for W rows (64 used)
#define LDS_XR  0
#define LDS_XI  (64 * SX_)                  // 2560
#define LDS_WR  (2 * 64 * SX_)              // 5120
#define LDS_WI  (2 * 64 * SX_ + 64 * SW_)   // 9728
#define LDS_TOT (2 * 64 * SX_ + 2 * 64 * SW_) // 14336 floats = 56 KB

// ---------------------------------------------------------------------------
// Async global->LDS copy of 16 bytes (CDNA5 GLOBAL_LOAD_ASYNC_TO_LDS_B128,
// tracked by ASYNCcnt). lds_byte_off is the per-lane LDS byte address
// (low 32 bits of the flat shared-aperture address).
// ---------------------------------------------------------------------------
__device__ __forceinline__ void async_copy_b128(unsigned lds_byte_off,
                                                const float* gsrc) {
    asm volatile("global_load_async_to_lds_b128 %0, %1, off"
                 :: "v"(lds_byte_off), "v"(gsrc)
                 : "memory");
}

__device__ __forceinline__ void wait_async_zero() {
    asm volatile("s_wait_asynccnt 0x0" ::: "memory");
}

// ---------------------------------------------------------------------------
// In-place 128-point radix-2 DIT FFT on LDS arrays. 64 worker threads per row
// (t in [0,64)). sign = -1 forward, +1 inverse. Caller syncs before entry.
// ---------------------------------------------------------------------------
__device__ __forceinline__ void fft128(float* re, float* im, int t, float sign) {
    // bit-reverse permutation (7 bits); pairs are disjoint -> race-free
#pragma unroll
    for (int q = 0; q < 2; ++q) {
        int j = t + 64 * q;
        int r = (int)(__brev((unsigned)j) >> 25);
        if (r > j) {
            float tr = re[j]; re[j] = re[r]; re[r] = tr;
            float ti = im[j]; im[j] = im[r]; im[r] = ti;
        }
    }
    __syncthreads();
#pragma unroll
    for (int s = 1; s <= 7; ++s) {
        int half = 1 << (s - 1);
        int j    = t & (half - 1);
        int base = ((t >> (s - 1)) << s) | j;
        float ang = sign * 3.14159265358979323846f * (float)j / (float)half;
        float sn, cs;
        __sincosf(ang, &sn, &cs);
        float vr = re[base + half], vi = im[base + half];
        float tr = vr * cs - vi * sn;
        float ti = vr * sn + vi * cs;
        float ur = re[base], ui = im[base];
        re[base]        = ur + tr;  im[base]        = ui + ti;
        re[base + half] = ur - tr;  im[base + half] = ui - ti;
        __syncthreads();
    }
}

// ---------------------------------------------------------------------------
// Stage 0: transpose weights (h,i,o,mx,y) -> per-frequency-point (p, i, o)
// so WMMA B-fragment loads are contiguous in o.
// ---------------------------------------------------------------------------
__global__ void wt_transpose_kernel(const float* __restrict__ wr,
                                    const float* __restrict__ wi,
                                    float* __restrict__ Wtr,
                                    float* __restrict__ Wti) {
    long idx = (long)blockIdx.x * 256 + threadIdx.x;
    const long total = (long)NPTS_ * C_IN * C_OUT;
    if (idx >= total) return;
    int  o  = (int)(idx % C_OUT);
    long r1 = idx / C_OUT;
    int  i  = (int)(r1 % C_IN);
    long p  = r1 / C_IN;
    int  y   = (int)(p % WY_);
    int  row = (int)(p / WY_);
    int  h   = row >> 4;          // head
    int  mx  = row & 15;          // x-mode within head
    long src = ((((long)h * C_IN + i) * C_OUT + o) * MPX_ + mx) * WY_ + y;
    Wtr[idx] = wr[src];
    Wti[idx] = wi[src];
}

// ---------------------------------------------------------------------------
// Stage 1: rfft along W for every (b,c,h) row. Real input -> 65 complex bins.
// Block: (64, 4) = 4 rows per block.
// ---------------------------------------------------------------------------
__global__ void __launch_bounds__(256)
rfft_w_kernel(const float* __restrict__ x,
              float* __restrict__ S1r, float* __restrict__ S1i) {
    __shared__ float re[4][128];
    __shared__ float im[4][128];
    int t  = threadIdx.x;
    int ry = threadIdx.y;
    long row = (long)blockIdx.x * 4 + ry;     // B*C*H rows total (exact)
    const float* xp = x + row * W_DIM;
    re[ry][t]      = xp[t];
    re[ry][t + 64] = xp[t + 64];
    im[ry][t]      = 0.0f;
    im[ry][t + 64] = 0.0f;
    __syncthreads();
    fft128(re[ry], im[ry], t, -1.0f);
    float* orp = S1r + row * WY_;
    float* oip = S1i + row * WY_;
    orp[t] = re[ry][t] * INV_SQRT_128;
    oip[t] = im[ry][t] * INV_SQRT_128;
    if (t == 0) {
        orp[64] = re[ry][64] * INV_SQRT_128;
        oip[64] = im[ry][64] * INV_SQRT_128;
    }
}

// ---------------------------------------------------------------------------
// Stage 2: complex FFT along H for every (b,c,y) column. Keep rows 0..63,
// store transposed as (p=row*WY+y, i=c, b) for coalesced WMMA A loads.
// ---------------------------------------------------------------------------
__global__ void __launch_bounds__(256)
fft_h_kernel(const float* __restrict__ S1r, const float* __restrict__ S1i,
             float* __restrict__ Xtr, float* __restrict__ Xti) {
    __shared__ float re[4][128];
    __shared__ float im[4][128];
    int t  = threadIdx.x;
    int ry = threadIdx.y;
    long col = (long)blockIdx.x * 4 + ry;     // B*C*WY columns (exact multiple of 4)
    int  y  = (int)(col % WY_);
    long bc = col / WY_;                      // b*C_IN + c
    int  c  = (int)(bc % C_IN);
    int  b  = (int)(bc / C_IN);
    const float* ipr = S1r + bc * (long)(H_DIM * WY_) + y;
    const float* ipi = S1i + bc * (long)(H_DIM * WY_) + y;
    re[ry][t]      = ipr[(long)t * WY_];
    re[ry][t + 64] = ipr[(long)(t + 64) * WY_];
    im[ry][t]      = ipi[(long)t * WY_];
    im[ry][t + 64] = ipi[(long)(t + 64) * WY_];
    __syncthreads();
    fft128(re[ry], im[ry], t, -1.0f);
    // rows 0..63 only; thread t owns output row t
    long p   = (long)t * WY_ + y;
    long dst = (p * C_IN + c) * BATCH + b;
    Xtr[dst] = re[ry][t] * INV_SQRT_128;
    Xti[dst] = im[ry][t] * INV_SQRT_128;
}

// ---------------------------------------------------------------------------
// Stage 3: per-frequency-point complex GEMM via V_WMMA_F32_16X16X4_F32.
// out[b,o] = sum_i X[b,i] * W[i,o]   (complex, 32x64 @ 64x64)
// One block per point; the X (16 KB) and W (32 KB) panels are staged into
// LDS once per workgroup via GLOBAL_LOAD_ASYNC_TO_LDS_B128 (ASYNCcnt), with
// bank-conflict-free padded strides; 8 waves = 2 (M) x 4 (N) 16x16 tiles.
// ---------------------------------------------------------------------------
__global__ void __launch_bounds__(256)
spectral_wmma_kernel(const float* __restrict__ Xtr, const float* __restrict__ Xti,
                     const float* __restrict__ Wtr, const float* __restrict__ Wti,
                     float* __restrict__ Ytr, float* __restrict__ Yti) {
    __shared__ float lds[LDS_TOT];
    unsigned ldsbase = (unsigned)(uintptr_t)lds;  // low 32 bits = LDS byte offset

    int p    = blockIdx.x;                 // 0..4159
    int tid  = threadIdx.x;
    int wave = tid >> 5;                   // 0..7
    int lane = tid & 31;
    int tm   = wave >> 2;                  // M tile (batch 0..15 / 16..31)
    int tn   = wave & 3;                   // N tile (out-ch block)
    int hl   = lane & 15;
    int hi   = lane >> 4;

    const float* Xr = Xtr + (long)p * C_IN * BATCH;
    const float* Xi = Xti + (long)p * C_IN * BATCH;
    const float* Wr = Wtr + (long)p * C_IN * C_OUT;
    const float* Wi = Wti + (long)p * C_IN * C_OUT;

    // ---- async stage: X rows (32 floats -> stride SX_), W rows (64 -> SW_) --
#pragma unroll
    for (int q = 0; q < 2; ++q) {          // 512 b128 chunks per X plane
        int c = tid + q * 256;
        int i = c >> 3;                    // row
        int w = c & 7;                     // 16B chunk within row
        async_copy_b128(ldsbase + (unsigned)(LDS_XR + i * SX_ + w * 4) * 4u,
                        Xr + i * 32 + w * 4);
        async_copy_b128(ldsbase + (unsigned)(LDS_XI + i * SX_ + w * 4) * 4u,
                        Xi + i * 32 + w * 4);
    }
#pragma unroll
    for (int q = 0; q < 4; ++q) {          // 1024 b128 chunks per W plane
        int c = tid + q * 256;
        int i = c >> 4;
        int w = c & 15;
        async_copy_b128(ldsbase + (unsigned)(LDS_WR + i * SW_ + w * 4) * 4u,
                        Wr + i * 64 + w * 4);
        async_copy_b128(ldsbase + (unsigned)(LDS_WI + i * SW_ + w * 4) * 4u,
                        Wi + i * 64 + w * 4);
    }
    wait_async_zero();                     // s_wait_asynccnt 0 (own wave's copies)
    __syncthreads();                       // all waves' copies visible

    int bidx = tm * 16 + hl;               // batch row handled by this lane (A side)
    int oidx = tn * 16 + hl;               // out-channel col handled by this lane (B side)

    v8f aRR = {}; v8f aII = {}; v8f aRI = {}; v8f aIR = {};

    for (int k = 0; k < 16; ++k) {
        int i0 = k * 4 + 2 * hi;           // K pair for this lane half
        v2f ar, ai, br, bi;
        // A fragment (16x4 of X): VGPR0 = K even, VGPR1 = K odd per lane half
        ar.x = lds[LDS_XR + i0 * SX_ + bidx];
        ar.y = lds[LDS_XR + (i0 + 1) * SX_ + bidx];
        ai.x = lds[LDS_XI + i0 * SX_ + bidx];
        ai.y = lds[LDS_XI + (i0 + 1) * SX_ + bidx];
        // B fragment (4x16 of W): rows striped across lanes within each VGPR
        br.x = lds[LDS_WR + i0 * SW_ + oidx];
        br.y = lds[LDS_WR + (i0 + 1) * SW_ + oidx];
        bi.x = lds[LDS_WI + i0 * SW_ + oidx];
        bi.y = lds[LDS_WI + (i0 + 1) * SW_ + oidx];

        aRR = __builtin_amdgcn_wmma_f32_16x16x4_f32(false, ar, false, br, (short)0, aRR, false, false);
        aII = __builtin_amdgcn_wmma_f32_16x16x4_f32(false, ai, false, bi, (short)0, aII, false, false);
        aRI = __builtin_amdgcn_wmma_f32_16x16x4_f32(false, ar, false, bi, (short)0, aRI, false, false);
        aIR = __builtin_amdgcn_wmma_f32_16x16x4_f32(false, ai, false, br, (short)0, aIR, false, false);
    }

    // Epilogue: complex combine + store. Y layout (p, b, o), coalesced in o.
    float* Yr = Ytr + (long)p * BATCH * C_OUT;
    float* Yi = Yti + (long)p * BATCH * C_OUT;
#pragma unroll
    for (int r = 0; r < 8; ++r) {
        int b = tm * 16 + 8 * hi + r;      // C/D layout: VGPR r -> M=r (lanes<16) / M=r+8
        long dst = (long)b * C_OUT + oidx;
        Yr[dst] = aRR[r] - aII[r];
        Yi[dst] = aRI[r] + aIR[r];
    }
}

// ---------------------------------------------------------------------------
// Stage 4: inverse complex FFT along H; input rows 64..127 are implicit zeros.
// Columns are (b,o,y). Output Z in (b,o,h,y) layout (reuses S1 scratch).
// ---------------------------------------------------------------------------
__global__ void __launch_bounds__(256)
ifft_h_kernel(const float* __restrict__ Ytr, const float* __restrict__ Yti,
              float* __restrict__ Zr, float* __restrict__ Zi) {
    __shared__ float re[4][128];
    __shared__ float im[4][128];
    int t  = threadIdx.x;
    int ry = threadIdx.y;
    long col = (long)blockIdx.x * 4 + ry;     // B*C_OUT*WY columns
    int  y  = (int)(col % WY_);
    long bo = col / WY_;                      // b*C_OUT + o
    int  o  = (int)(bo % C_OUT);
    int  b  = (int)(bo / C_OUT);
    // rows 0..63 from spectral output (p = h*WY + y, layout (p,b,o))
    long src = (((long)t * WY_ + y) * BATCH + b) * C_OUT + o;
    re[ry][t]      = Ytr[src];
    im[ry][t]      = Yti[src];
    re[ry][t + 64] = 0.0f;                    // zero-padded rows 64..127
    im[ry][t + 64] = 0.0f;
    __syncthreads();
    fft128(re[ry], im[ry], t, +1.0f);
    float* zr = Zr + bo * (long)(H_DIM * WY_);
    float* zi = Zi + bo * (long)(H_DIM * WY_);
    zr[(long)t * WY_ + y]        = re[ry][t] * INV_SQRT_128;
    zi[(long)t * WY_ + y]        = im[ry][t] * INV_SQRT_128;
    zr[(long)(t + 64) * WY_ + y] = re[ry][t + 64] * INV_SQRT_128;
    zi[(long)(t + 64) * WY_ + y] = im[ry][t + 64] * INV_SQRT_128;
}

// ---------------------------------------------------------------------------
// Stage 5: Hermitian-extended inverse FFT along W, take real part, add bias.
// ---------------------------------------------------------------------------
__global__ void __launch_bounds__(256)
irfft_w_kernel(const float* __restrict__ Zr, const float* __restrict__ Zi,
               const float* __restrict__ bias, float* __restrict__ out) {
    __shared__ float re[4][128];
    __shared__ float im[4][128];
    int t  = threadIdx.x;
    int ry = threadIdx.y;
    long row = (long)blockIdx.x * 4 + ry;     // B*C_OUT*H rows
    long bo  = row / H_DIM;
    int  o   = (int)(bo % C_OUT);
    const float* zr = Zr + row * WY_;
    const float* zi = Zi + row * WY_;
    // k = t (0..63) directly; k = t+64 via Hermitian symmetry X[k]=conj(X[128-k])
    re[ry][t] = zr[t];
    im[ry][t] = zi[t];
    if (t == 0) {
        re[ry][64] = zr[64];
        im[ry][64] = zi[64];
    } else {
        re[ry][t + 64] =  zr[64 - t];
        im[ry][t + 64] = -zi[64 - t];
    }
    __syncthreads();
    fft128(re[ry], im[ry], t, +1.0f);
    float bv = bias[o];
    float* op = out + row * W_DIM;
    op[t]      = re[ry][t]      * INV_SQRT_128 + bv;
    op[t + 64] = re[ry][t + 64] * INV_SQRT_128 + bv;
}

// ---------------------------------------------------------------------------
extern "C" void kernel_launch(void* const* d_in, const int* in_sizes, int n_in,
                              void* d_out, int out_size, void* d_ws, size_t ws_size,
                              hipStream_t stream) {
    const float* x    = (const float*)d_in[0];
    const float* wr   = (const float*)d_in[1];
    const float* wi   = (const float*)d_in[2];
    const float* bias = (const float*)d_in[3];
    float* out = (float*)d_out;
    float* ws  = (float*)d_ws;

    const size_t R1 = (size_t)BATCH * C_IN * H_DIM * WY_;   // 17,039,360  rfft-W output
    const size_t R2 = (size_t)NPTS_ * C_IN * BATCH;         //  8,519,680  transposed spectra
    const size_t R3 = (size_t)NPTS_ * C_IN * C_OUT;         // 17,039,360  transposed weights

    float* S1r = ws;                float* S1i = S1r + R1;
    float* Xtr = S1i + R1;          float* Xti = Xtr + R2;
    float* Wtr = Xti + R2;          float* Wti = Wtr + R3;
    float* Ytr = Wti + R3;          float* Yti = Ytr + R2;
    float* Zr  = S1r;               float* Zi  = S1i;       // reuse S1 after stage 2

    dim3 fftBlk(64, 4, 1);

    wt_transpose_kernel<<<(unsigned)((R3 + 255) / 256), 256, 0, stream>>>(wr, wi, Wtr, Wti);
    rfft_w_kernel<<<(BATCH * C_IN * H_DIM) / 4, fftBlk, 0, stream>>>(x, S1r, S1i);
    fft_h_kernel<<<(BATCH * C_IN * WY_) / 4, fftBlk, 0, stream>>>(S1r, S1i, Xtr, Xti);
    spectral_wmma_kernel<<<NPTS_, 256, 0, stream>>>(Xtr, Xti, Wtr, Wti, Ytr, Yti);
    ifft_h_kernel<<<(BATCH * C_OUT * WY_) / 4, fftBlk, 0, stream>>>(Ytr, Yti, Zr, Zi);
    irfft_w_kernel<<<(BATCH * C_OUT * H_DIM) / 4, fftBlk, 0, stream>>>(Zr, Zi, bias, out);
}